// HGNN_72730976190574
// MI455X (gfx1250) — compile-verified
//
#include <hip/hip_runtime.h>
#include <hip/hip_bf16.h>

typedef float v2f __attribute__((ext_vector_type(2)));
typedef float v8f __attribute__((ext_vector_type(8)));

// -------------------------------------------------------------------------
// grid-stride zero (graph-capture-safe replacement for memset)
// -------------------------------------------------------------------------
__global__ void zero_f32(float* __restrict__ p, size_t n) {
    size_t i = (size_t)blockIdx.x * blockDim.x + threadIdx.x;
    size_t stride = (size_t)gridDim.x * blockDim.x;
    for (; i < n; i += stride) p[i] = 0.0f;
}

// -------------------------------------------------------------------------
// one wave32 per edge: gather src row, atomicAdd into dst row (+count)
// -------------------------------------------------------------------------
__global__ __launch_bounds__(256)
void scatter_edges(const float* __restrict__ xsrc, const int* __restrict__ edge,
                   float* __restrict__ agg, float* __restrict__ cnt,
                   int E, int F) {
    int w    = (int)(((size_t)blockIdx.x * blockDim.x + threadIdx.x) >> 5);
    int lane = threadIdx.x & 31;
    if (w >= E) return;
    int s = edge[w];        // src node
    int d = edge[E + w];    // dst node
    const float* src = xsrc + (size_t)s * F;
    float*       dst = agg  + (size_t)d * F;
    for (int f = lane; f < F; f += 32)
        atomicAdd(&dst[f], src[f]);
    if (lane == 0) atomicAdd(&cnt[d], 1.0f);
}

// -------------------------------------------------------------------------
// out[n,128] = act( (agg/max(cnt,1)) @ Wl  +  bias  +  xd @ Wr )
// f32 WMMA 16x16x4: one 16x16 tile per wave, 8 waves cover 16x128 rows/block
//   A lane layout : row = lane%16, k = (lane/16)*2 + {0,1}
//   B lane layout : col = lane%16, k = (lane/16)*2 + {0,1}
//   C/D layout    : col = lane%16, row = (lane/16)*8 + vgpr
// k_agg==0 skips the aggregate path (used for the readout GEMM).
// -------------------------------------------------------------------------
__global__ __launch_bounds__(256)
void sage_gemm_wmma(const float* __restrict__ agg, const float* __restrict__ cnt,
                    int k_agg,
                    const float* __restrict__ xd, int k_dst,
                    const float* __restrict__ Wl, const float* __restrict__ Wr,
                    const float* __restrict__ bias,
                    float* __restrict__ out, int n, int do_relu) {
    const int lane = threadIdx.x & 31;
    const int wave = threadIdx.x >> 5;
    const int c0   = wave << 4;            // column tile base (0..112)
    const int r0   = blockIdx.x << 4;      // row tile base
    const int rl   = lane & 15;
    const int kh   = (lane >> 4) << 1;     // 0 or 2
    int rc = r0 + rl;
    if (rc > n - 1) rc = n - 1;            // clamp loads; stores masked below
    const int col = c0 + rl;

    v8f acc = {};

    if (k_agg > 0) {
        float c   = cnt[rc];
        float inv = 1.0f / (c > 1.0f ? c : 1.0f);
        const float* arow = agg + (size_t)rc * k_agg;
        for (int k0 = 0; k0 < k_agg; k0 += 4) {
            v2f a, b;
            a.x = arow[k0 + kh] * inv;
            a.y = arow[k0 + kh + 1] * inv;
            b.x = Wl[(size_t)(k0 + kh) * 128 + col];
            b.y = Wl[(size_t)(k0 + kh + 1) * 128 + col];
            acc = __builtin_amdgcn_wmma_f32_16x16x4_f32(
                      false, a, false, b, (short)0, acc, false, false);
        }
    }
    if (k_dst > 0) {
        const float* xrow = xd + (size_t)rc * k_dst;
        for (int k0 = 0; k0 < k_dst; k0 += 4) {
            v2f a, b;
            a.x = xrow[k0 + kh];
            a.y = xrow[k0 + kh + 1];
            b.x = Wr[(size_t)(k0 + kh) * 128 + col];
            b.y = Wr[(size_t)(k0 + kh + 1) * 128 + col];
            acc = __builtin_amdgcn_wmma_f32_16x16x4_f32(
                      false, a, false, b, (short)0, acc, false, false);
        }
    }

    const float bv    = bias[col];
    const int   mbase = (lane >> 4) << 3;  // 0 or 8
#pragma unroll
    for (int v = 0; v < 8; ++v) {
        int   row = r0 + mbase + v;
        float val = acc[v] + bv;
        if (do_relu && val < 0.0f) val = 0.0f;
        if (row < n) out[(size_t)row * 128 + col] = val;
    }
}

// -------------------------------------------------------------------------
// column mean of z[n,128] -> outv[128]   (one block per column)
// -------------------------------------------------------------------------
__global__ void col_mean(const float* __restrict__ z, float* __restrict__ outv, int n) {
    __shared__ float red[256];
    int c = blockIdx.x;
    float s = 0.0f;
    for (int r = threadIdx.x; r < n; r += blockDim.x)
        s += z[(size_t)r * 128 + c];
    red[threadIdx.x] = s;
    __syncthreads();
    for (int off = blockDim.x >> 1; off > 0; off >>= 1) {
        if ((int)threadIdx.x < off) red[threadIdx.x] += red[threadIdx.x + off];
        __syncthreads();
    }
    if (threadIdx.x == 0) outv[c] = red[0] / (float)n;
}

// -------------------------------------------------------------------------
// out[8] = softmax(fvec[256] @ Wfc[256,8] + bfc[8])
// -------------------------------------------------------------------------
__global__ void head_softmax(const float* __restrict__ fvec, const float* __restrict__ Wfc,
                             const float* __restrict__ bfc, float* __restrict__ out) {
    __shared__ float v[256];
    __shared__ float lg[8];
    for (int i = threadIdx.x; i < 256; i += blockDim.x) v[i] = fvec[i];
    __syncthreads();
    if (threadIdx.x < 8) {
        float a = bfc[threadIdx.x];
        for (int k = 0; k < 256; ++k) a += v[k] * Wfc[k * 8 + threadIdx.x];
        lg[threadIdx.x] = a;
    }
    __syncthreads();
    if (threadIdx.x == 0) {
        float mx = lg[0];
        for (int o = 1; o < 8; ++o) mx = fmaxf(mx, lg[o]);
        float e[8], s = 0.0f;
        for (int o = 0; o < 8; ++o) { e[o] = expf(lg[o] - mx); s += e[o]; }
        for (int o = 0; o < 8; ++o) out[o] = e[o] / s;
    }
}

// -------------------------------------------------------------------------
extern "C" void kernel_launch(void* const* d_in, const int* in_sizes, int n_in,
                              void* d_out, int out_size, void* d_ws, size_t ws_size,
                              hipStream_t stream) {
    const int NA = 50000, NB = 50000, HID = 128, FA = 128, FB = 64;
    const int E = 1600000;

    const float* x_a   = (const float*)d_in[0];
    const float* x_b   = (const float*)d_in[1];
    const int*   e_ab  = (const int*)d_in[2];
    const int*   e_ba  = (const int*)d_in[3];
    const float* W0abl = (const float*)d_in[4];   // (128,128)
    const float* b0ab  = (const float*)d_in[5];
    const float* W0abr = (const float*)d_in[6];   // (64,128)
    const float* W0bal = (const float*)d_in[7];   // (64,128)
    const float* b0ba  = (const float*)d_in[8];
    const float* W0bar = (const float*)d_in[9];   // (128,128)
    const float* W1abl = (const float*)d_in[10];
    const float* b1ab  = (const float*)d_in[11];
    const float* W1abr = (const float*)d_in[12];
    const float* W1bal = (const float*)d_in[13];
    const float* b1ba  = (const float*)d_in[14];
    const float* W1bar = (const float*)d_in[15];
    const float* Wn    = (const float*)d_in[16];
    const float* bn    = (const float*)d_in[17];
    const float* Wfc   = (const float*)d_in[18];
    const float* bfc   = (const float*)d_in[19];
    float* out = (float*)d_out;

    // ---- workspace carve-up (all f32) ------------------------------------
    float* ws    = (float*)d_ws;
    float* agg_a = ws;  ws += (size_t)NA * HID;
    float* agg_b = ws;  ws += (size_t)NB * HID;
    float* cnt_a = ws;  ws += NA;
    float* cnt_b = ws;  ws += NB;
    float* ha    = ws;  ws += (size_t)NA * HID;
    float* hb    = ws;  ws += (size_t)NB * HID;
    float* ha2   = ws;  ws += (size_t)NA * HID;
    float* hb2   = ws;  ws += (size_t)NB * HID;
    float* fvec  = ws;  ws += 256;
    float* z_a = agg_a;            // reuse agg buffers after layer 1
    float* z_b = agg_b;

    const size_t zeroN = (size_t)NA * HID + (size_t)NB * HID + NA + NB; // agg_a..cnt_b contiguous
    dim3 blk(256);
    const int zgrid = 2048;
    const int sgrid = (int)(((size_t)E * 32 + 255) / 256);
    const int ggA = (NA + 15) / 16, ggB = (NB + 15) / 16;

    // ---- layer 0 ---------------------------------------------------------
    zero_f32<<<zgrid, blk, 0, stream>>>(agg_a, zeroN);
    scatter_edges<<<sgrid, blk, 0, stream>>>(x_b, e_ba, agg_a, cnt_a, E, FB);
    scatter_edges<<<sgrid, blk, 0, stream>>>(x_a, e_ab, agg_b, cnt_b, E, FA);
    sage_gemm_wmma<<<ggA, blk, 0, stream>>>(agg_a, cnt_a, FB, x_a, FA,
                                            W0bal, W0bar, b0ba, ha, NA, 1);
    sage_gemm_wmma<<<ggB, blk, 0, stream>>>(agg_b, cnt_b, FA, x_b, FB,
                                            W0abl, W0abr, b0ab, hb, NB, 1);

    // ---- layer 1 ---------------------------------------------------------
    zero_f32<<<zgrid, blk, 0, stream>>>(agg_a, zeroN);
    scatter_edges<<<sgrid, blk, 0, stream>>>(hb, e_ba, agg_a, cnt_a, E, HID);
    scatter_edges<<<sgrid, blk, 0, stream>>>(ha, e_ab, agg_b, cnt_b, E, HID);
    sage_gemm_wmma<<<ggA, blk, 0, stream>>>(agg_a, cnt_a, HID, ha, HID,
                                            W1bal, W1bar, b1ba, ha2, NA, 1);
    sage_gemm_wmma<<<ggB, blk, 0, stream>>>(agg_b, cnt_b, HID, hb, HID,
                                            W1abl, W1abr, b1ab, hb2, NB, 1);

    // ---- readout: z = relu(h @ Wn + bn); column means; softmax head ------
    sage_gemm_wmma<<<ggA, blk, 0, stream>>>(nullptr, nullptr, 0, ha2, HID,
                                            Wn, Wn, bn, z_a, NA, 1);
    sage_gemm_wmma<<<ggB, blk, 0, stream>>>(nullptr, nullptr, 0, hb2, HID,
                                            Wn, Wn, bn, z_b, NB, 1);
    col_mean<<<128, blk, 0, stream>>>(z_a, fvec, NA);
    col_mean<<<128, blk, 0, stream>>>(z_b, fvec + 128, NB);
    head_softmax<<<1, 64, 0, stream>>>(fvec, Wfc, bfc, out);
}